// MyModel_61933428413849
// MI455X (gfx1250) — compile-verified
//
#include <hip/hip_runtime.h>
#include <stdint.h>

// Beta(a,b) sampling via two Marsaglia-Tsang Gamma draws, S=10 samples of a
// (1024,4096) parameter grid. Pure streaming workload: 167.8MB write-dominated
// => HBM-bound (~8.6us floor @ 23.3TB/s). CDNA5 paths used: async global->LDS
// staging of parameter tiles (ASYNCcnt), wave32 blocking, NT b128 stores.

#define NSAMP 10
#define TILE_ELEMS 1024   // 256 threads * 4 floats

typedef float v4f __attribute__((ext_vector_type(4)));

__device__ __forceinline__ uint64_t splitmix64(uint64_t z) {
    z += 0x9E3779B97F4A7C15ULL;
    z = (z ^ (z >> 30)) * 0xBF58476D1CE4E5B9ULL;
    z = (z ^ (z >> 27)) * 0x94D049BB133111EBULL;
    return z ^ (z >> 31);
}

__device__ __forceinline__ uint32_t pcg32(uint64_t& st) {
    uint64_t old = st;
    st = old * 6364136223846793005ULL + 1442695040888963407ULL;
    uint32_t xs  = (uint32_t)(((old >> 18) ^ old) >> 27);
    uint32_t rot = (uint32_t)(old >> 59);
    return (xs >> rot) | (xs << ((32u - rot) & 31u));
}

__device__ __forceinline__ float uniform01(uint64_t& st) {
    // (0,1): 24-bit mantissa, never exactly 0 or 1
    return ((float)(pcg32(st) >> 8) + 0.5f) * (1.0f / 16777216.0f);
}

__device__ __forceinline__ float normal01(uint64_t& st) {
    float u1 = uniform01(st);
    float u2 = uniform01(st);
    float r = __builtin_sqrtf(-2.0f * __logf(u1));
    return r * __cosf(6.28318530717958647692f * u2);
}

// Marsaglia-Tsang rejection sampler for Gamma(alpha, 1), general alpha > 0.
__device__ __forceinline__ float gamma_sample(uint64_t& st, float alpha) {
    float boost = 1.0f;
    if (alpha < 1.0f) {                       // boost: G(a) = G(a+1) * U^(1/a)
        boost = __powf(uniform01(st), __frcp_rn(alpha));
        alpha += 1.0f;
    }
    const float d = alpha - 0.333333333333f;
    const float c = __frsqrt_rn(9.0f * d);    // 1/(3*sqrt(d))
    for (;;) {
        float x = normal01(st);
        float t = __fmaf_rn(c, x, 1.0f);
        if (t <= 0.0f) continue;
        float v  = t * t * t;
        float u  = uniform01(st);
        float x2 = x * x;
        float dv = d * v;
        if (u < 1.0f - 0.0331f * x2 * x2) return dv * boost;   // squeeze: no log
        if (__logf(u) < 0.5f * x2 + d - dv + d * __logf(v)) return dv * boost;
    }
}

__global__ __launch_bounds__(256) void beta_rsample_kernel(
    const float* __restrict__ a, const float* __restrict__ b,
    float* __restrict__ out, long long n_elem)
{
    __shared__ float ldsA[TILE_ELEMS];
    __shared__ float ldsB[TILE_ELEMS];

    const int       tid   = threadIdx.x;
    const long long tile0 = (long long)blockIdx.x * TILE_ELEMS;
    const long long base  = tile0 + (long long)tid * 4;
    if (base + 3 >= n_elem) return;   // full tiles only (n is a multiple of 1024)

    // ---- CDNA5 async DMA: stage this block's a/b tiles into LDS ----------
    // GVS mode: SGPR 64-bit base + per-lane 32-bit VGPR byte offset.
    // Low 32 bits of a generic shared pointer are the LDS byte offset
    // (ISA 10.2 aperture mapping).
    const float* ga = a + tile0;
    const float* gb = b + tile0;
    uint32_t voff = (uint32_t)(tid * 16);
    uint32_t ldsa = (uint32_t)(uintptr_t)(&ldsA[0]) + voff;
    uint32_t ldsb = (uint32_t)(uintptr_t)(&ldsB[0]) + voff;
    asm volatile(
        "global_load_async_to_lds_b128 %0, %2, %3\n\t"
        "global_load_async_to_lds_b128 %1, %2, %4"
        :
        : "v"(ldsa), "v"(ldsb), "v"(voff), "s"(ga), "s"(gb)
        : "memory");

    // RNG init overlaps with the async DMA. Seed depends only on the element
    // index -> deterministic across calls / graph replays.
    uint64_t st = splitmix64(0xC0FFEE1234ULL ^ (uint64_t)(base + 1));

    // Each lane reads back exactly the 16B it requested: same-wave ASYNCcnt
    // ordering is sufficient, no workgroup barrier needed.
    asm volatile("s_wait_asynccnt 0x0" ::: "memory");

    v4f av = *(const v4f*)&ldsA[tid * 4];
    v4f bv = *(const v4f*)&ldsB[tid * 4];

    // a,b live in registers across all S samples -> parameters read from HBM
    // exactly once. Output written once, non-temporally (streaming 168MB).
    #pragma unroll
    for (int s = 0; s < NSAMP; ++s) {
        v4f o;
        #pragma unroll
        for (int j = 0; j < 4; ++j) {
            float g1 = gamma_sample(st, av[j]);
            float g2 = gamma_sample(st, bv[j]);
            o[j] = g1 * __frcp_rn(g1 + g2);
        }
        __builtin_nontemporal_store(o, (v4f*)(out + (long long)s * n_elem + base));
    }
}

extern "C" void kernel_launch(void* const* d_in, const int* in_sizes, int n_in,
                              void* d_out, int out_size, void* d_ws, size_t ws_size,
                              hipStream_t stream) {
    // setup_inputs order: d_in[0]=dummy_input(1), d_in[1]=a, d_in[2]=b
    const float* a = (const float*)d_in[1];
    const float* b = (const float*)d_in[2];
    float* out = (float*)d_out;
    long long n_elem = (long long)in_sizes[1];            // 1024*4096
    int blocks = (int)((n_elem + TILE_ELEMS - 1) / TILE_ELEMS);
    beta_rsample_kernel<<<dim3(blocks), dim3(256), 0, stream>>>(a, b, out, n_elem);
}